// SingleSideChamferLoss_Brute_2370821948146
// MI455X (gfx1250) — compile-verified
//
#include <hip/hip_runtime.h>
#include <math.h>

typedef __attribute__((ext_vector_type(2))) float v2f;
typedef __attribute__((ext_vector_type(8))) float v8f;

#define M_NODES   1024
#define N_POINTS  16384
#define BATCH     8
#define WAVES_PER_BLOCK 8
#define COLS_PER_WAVE   16
#define COLS_PER_BLOCK  (WAVES_PER_BLOCK * COLS_PER_WAVE)   // 128
#define M_CHUNKS  (M_NODES / 16)                            // 64

// One workgroup: batch b, side z, 128 columns (8 waves x 16 cols).
// LDS holds all M=1024 nodes in augmented form: row m = {-2x0,-2x1,-2x2,|x|^2}.
// Each wave: 64x V_WMMA_F32_16X16X4_F32 producing 16x16 score tiles
// D[m][n] = |x_m|^2 - 2 x_m . y_n  (B operand per column = {y0,y1,y2,1}).
// min over m folds as: per-lane fmin over 8 D regs, then shfl_xor(16) merge.
__global__ __launch_bounds__(256)
void chamfer_min_kernel(const float* __restrict__ node_pos,
                        const float* __restrict__ pts_pos,
                        const float* __restrict__ node_anc,
                        const float* __restrict__ pts_anc,
                        float* __restrict__ partials)
{
    __shared__ float ldsA[M_NODES * 4];
    __shared__ float wave_part[WAVES_PER_BLOCK];

    const int side = blockIdx.z;
    const int b    = blockIdx.y;
    const float* __restrict__ node = side ? node_anc : node_pos;
    const float* __restrict__ pts  = side ? pts_anc  : pts_pos;

    const int tid = threadIdx.x;

    // Stage 1: preprocess nodes of this batch into LDS (augmented rows).
    for (int m = tid; m < M_NODES; m += 256) {
        const float* xp = node + ((size_t)b * M_NODES + m) * 3;
        float x0 = xp[0], x1 = xp[1], x2 = xp[2];
        ldsA[m * 4 + 0] = -2.0f * x0;
        ldsA[m * 4 + 1] = -2.0f * x1;
        ldsA[m * 4 + 2] = -2.0f * x2;
        ldsA[m * 4 + 3] = x0 * x0 + x1 * x1 + x2 * x2;
    }
    __syncthreads();

    const int wave = tid >> 5;
    const int lane = tid & 31;
    const int half = lane >> 4;   // 0: lanes 0-15 (K=0,1), 1: lanes 16-31 (K=2,3)
    const int l15  = lane & 15;

    // Stage 2: B fragment for this wave's 16 columns.
    const int n = blockIdx.x * COLS_PER_BLOCK + wave * COLS_PER_WAVE + l15;
    const float* yp = pts + ((size_t)b * N_POINTS + n) * 3;
    float y0 = yp[0], y1 = yp[1], y2 = yp[2];
    float ysq = y0 * y0 + y1 * y1 + y2 * y2;

    v2f bfrag;
    bfrag[0] = (half == 0) ? y0 : y2;
    bfrag[1] = (half == 0) ? y1 : 1.0f;

    float runmin = 3.4e38f;
    const float* arow = ldsA + half * 2;   // K offset 0 or 2 within each 16B row

    #pragma unroll 4
    for (int mc = 0; mc < M_CHUNKS; ++mc) {
        int m = mc * 16 + l15;
        v2f afrag = *(const v2f*)(arow + m * 4);   // ds_load_b64, conflict-free
        v8f c = {};
        v8f d = __builtin_amdgcn_wmma_f32_16x16x4_f32(
            /*neg_a=*/false, afrag, /*neg_b=*/false, bfrag,
            /*c_mod=*/(short)0, c, /*reuse_a=*/false, /*reuse_b=*/false);
        float t0 = fminf(d[0], d[1]);
        float t1 = fminf(d[2], d[3]);
        float t2 = fminf(d[4], d[5]);
        float t3 = fminf(d[6], d[7]);
        runmin = fminf(runmin, fminf(fminf(t0, t1), fminf(t2, t3)));
    }

    // Merge row halves: lanes 0-15 hold rows 0-7, lanes 16-31 rows 8-15 of same column.
    runmin = fminf(runmin, __shfl_xor(runmin, 16, 32));

    // Distance for this column (only lanes 0-15 carry a valid column result).
    float dist = (half == 0) ? sqrtf(fmaxf(runmin + ysq, 0.0f)) : 0.0f;

    // Deterministic wave sum (fixed shuffle tree).
    float s = dist;
    s += __shfl_xor(s, 1, 32);
    s += __shfl_xor(s, 2, 32);
    s += __shfl_xor(s, 4, 32);
    s += __shfl_xor(s, 8, 32);
    s += __shfl_xor(s, 16, 32);
    if (lane == 0) wave_part[wave] = s;
    __syncthreads();

    if (tid == 0) {
        float tot = 0.0f;
        #pragma unroll
        for (int w = 0; w < WAVES_PER_BLOCK; ++w) tot += wave_part[w];
        size_t wg = (size_t)blockIdx.z * gridDim.y * gridDim.x
                  + (size_t)blockIdx.y * gridDim.x + blockIdx.x;
        partials[wg] = tot;
    }
}

// Single-block, fixed-order final reduction -> deterministic scalar.
__global__ __launch_bounds__(256)
void chamfer_reduce_kernel(const float* __restrict__ partials, int n,
                           float* __restrict__ out)
{
    __shared__ float sh[256];
    float s = 0.0f;
    for (int i = threadIdx.x; i < n; i += 256) s += partials[i];
    sh[threadIdx.x] = s;
    __syncthreads();
    for (int off = 128; off > 0; off >>= 1) {
        if ((int)threadIdx.x < off) sh[threadIdx.x] += sh[threadIdx.x + off];
        __syncthreads();
    }
    if (threadIdx.x == 0)
        out[0] = sh[0] / (2.0f * (float)BATCH * (float)N_POINTS);
}

extern "C" void kernel_launch(void* const* d_in, const int* in_sizes, int n_in,
                              void* d_out, int out_size, void* d_ws, size_t ws_size,
                              hipStream_t stream) {
    const float* node_pos = (const float*)d_in[0];   // shifted_pos_points_c (B,M,3)
    const float* pts_pos  = (const float*)d_in[1];   // pos_points_f        (B,N,3)
    const float* node_anc = (const float*)d_in[2];   // shifted_anc_points_c(B,M,3)
    const float* pts_anc  = (const float*)d_in[3];   // anc_points_f        (B,N,3)
    float* out      = (float*)d_out;
    float* partials = (float*)d_ws;                  // 2*8*128 = 2048 floats (8 KB)

    dim3 grid(N_POINTS / COLS_PER_BLOCK, BATCH, 2);  // (128, 8, 2)
    chamfer_min_kernel<<<grid, 256, 0, stream>>>(node_pos, pts_pos, node_anc,
                                                 pts_anc, partials);

    int n_part = 2 * BATCH * (N_POINTS / COLS_PER_BLOCK);
    chamfer_reduce_kernel<<<1, 256, 0, stream>>>(partials, n_part, out);
}